// Tank4Dpl_81106162418000
// MI455X (gfx1250) — compile-verified
//
#include <hip/hip_runtime.h>
#include <hip/hip_bf16.h>

// Tank hydrological model: sequential scan over T=1096 steps, parallel over
// B=16384 basins. No matrix ops => no WMMA. CDNA5 path used: async
// global->LDS copies (ASYNCcnt) double-buffered ahead of a latency-bound
// serial VALU recurrence; nontemporal stores for the write-once outputs.

#define T_TOTAL 1096
#define TWARM   365
#define CHUNK   32                    // timesteps staged per async chunk
#define BLOCK   64                    // threads per block = 2 wave32
#define NCHUNK  ((T_TOTAL + CHUNK - 1) / CHUNK)   // 35 (34 full + 8-step tail)

__device__ __forceinline__ void async_ld_f2(const float2* g, const float2* lds) {
  // global_load_async_to_lds_b64: VDST = VGPR holding LDS byte offset,
  // VADDR = 64-bit global address pair. Low 32 bits of a generic pointer to
  // __shared__ are the LDS offset.
  unsigned lofs = (unsigned)(unsigned long long)lds;
  asm volatile("global_load_async_to_lds_b64 %0, %1, off"
               :: "v"(lofs), "v"(g) : "memory");
}

template <int N>
__device__ __forceinline__ void wait_async() {
  asm volatile("s_wait_asynccnt %0" :: "n"(N) : "memory");
}

// Clamp-free fast path: never called; issue_chunk issues every chunk. Global side strength-
// reduces to base + k*(B*8); LDS side is lbuf + k*512.
__device__ __forceinline__ void issue_chunk_full(const float2* __restrict__ gbase, // &pe2[t0*B + b]
                                                 const float2* lbuf) {             // &buf[pb][tid]
#pragma unroll
  for (int k = 0; k < CHUNK; ++k) {
    async_ld_f2(gbase, lbuf + k * BLOCK);
    gbase += 16384;   // placeholder; real stride patched by caller via pointer math
  }
}

// Generic issuer with tail clamping (only the last chunk needs it).
__device__ __forceinline__ void issue_chunk(const float2* __restrict__ gbase, // &pe2[b], t=0 row
                                            const float2* lbuf, int B, int t0,
                                            bool clamp_tail) {
  if (!clamp_tail) {
    const float2* g = gbase + (size_t)t0 * B;
#pragma unroll
    for (int k = 0; k < CHUNK; ++k) {
      async_ld_f2(g, lbuf + k * BLOCK);
      g += B;                           // constant +B*8 bytes per step
    }
  } else {
#pragma unroll
    for (int k = 0; k < CHUNK; ++k) {
      int tg = t0 + k;                      // pad tail with clamped (valid) addrs;
      if (tg > T_TOTAL - 1) tg = T_TOTAL - 1; // padded LDS slots are never read
      async_ld_f2(gbase + (size_t)tg * B, lbuf + k * BLOCK);
    }
  }
}

__global__ __launch_bounds__(BLOCK)
void tank_scan_kernel(const float* __restrict__ p_and_e,   // [T, B, 2]
                      const float* __restrict__ params,    // [B, 20] normalized
                      float* __restrict__ out,             // [731*B] q | [731*B] e
                      int B) {
  __shared__ float2 buf[2][CHUNK * BLOCK];   // 32 KB: double-buffered (p,e) tiles

  const int tid = threadIdx.x;
  const int b   = blockIdx.x * BLOCK + tid;
  if (b >= B) return;

  // ---- parameters: 20 floats per basin, 16B-aligned (80 bytes) -> 5x b128 ----
  constexpr float LO[20] = {0.5f, 1.0f, 1.0f, 0.01f, 0.01f, 0.01f, 0.01f, 0.01f,
                            0.0f, 0.0f, 0.01f, 0.01f, 0.0f, 0.01f, 0.0f, 0.01f,
                            0.001f, 0.01f, 0.01f, 0.0f};
  constexpr float HI[20] = {1.5f, 100.0f, 100.0f, 1.0f, 1.0f, 1.0f, 1.0f, 1.0f,
                            90.0f, 100.0f, 1.0f, 1.0f, 100.0f, 1.0f, 100.0f, 1.0f,
                            1.0f, 1.0f, 1.0f, 100.0f};
  float raw[20];
  {
    const float4* p4 = (const float4*)(params + (size_t)b * 20);
    float4 r0 = p4[0], r1 = p4[1], r2 = p4[2], r3 = p4[3], r4 = p4[4];
    raw[0]=r0.x; raw[1]=r0.y; raw[2]=r0.z; raw[3]=r0.w;
    raw[4]=r1.x; raw[5]=r1.y; raw[6]=r1.z; raw[7]=r1.w;
    raw[8]=r2.x; raw[9]=r2.y; raw[10]=r2.z; raw[11]=r2.w;
    raw[12]=r3.x; raw[13]=r3.y; raw[14]=r3.z; raw[15]=r3.w;
    raw[16]=r4.x; raw[17]=r4.y; raw[18]=r4.z; raw[19]=r4.w;
  }
  float ph[20];
#pragma unroll
  for (int i = 0; i < 20; ++i) ph[i] = fmaf(raw[i], HI[i] - LO[i], LO[i]);

  const float kc = ph[0], w1 = ph[1], w2 = ph[2], k1 = ph[3], k2 = ph[4];
  const float a0 = ph[5], b0 = ph[6], c0 = ph[7], h1 = ph[8], h2 = ph[9];
  const float a1 = ph[10], a2 = ph[11], h3 = ph[12], b1 = ph[13], h4 = ph[14];
  const float c1 = ph[15], d1 = ph[16], e1 = ph[17], e2 = ph[18], h = ph[19];

  // ---- hoist all divisions out of the serial loop ----
  const float dt      = 0.0005f;             // 0.5 * (1/1000)
  const float kw      = k2 / (w1 + w2);
  const float inv_e1  = 1.0f / e1;
  const float inv_e12 = 1.0f / (e1 + e2);
  const float ch      = e2 * h * inv_e12;    // e2*h/(e1+e2)
  const float ka      = inv_e1  - dt;        // (kk0 - dt) when x5 <  h
  const float kb      = inv_e12 - dt;        // (kk0 - dt) when x5 >= h
  const float da      = 1.0f / (inv_e1  + dt);
  const float db      = 1.0f / (inv_e12 + dt);

  // ---- state init ----
  float xf = 0.01f, xp = 0.01f, x2 = 0.01f, xs = 0.01f;
  float x3 = 0.01f, x4 = 0.01f, x5 = 0.01f, qs = 0.01f;

  const float2* gpe = ((const float2*)p_and_e) + b;
  float* __restrict__ qout = out;
  float* __restrict__ eout = out + (size_t)(T_TOTAL - TWARM) * B;

  // ---- software pipeline: async-prefetch chunk c+1 while computing chunk c ----
  issue_chunk(gpe, &buf[0][tid], B, 0, /*clamp_tail=*/false);

  for (int c = 0; c < NCHUNK; ++c) {
    if (c + 1 < NCHUNK) {
      issue_chunk(gpe, &buf[(c + 1) & 1][tid], B, (c + 1) * CHUNK,
                  /*clamp_tail=*/(c + 1 == NCHUNK - 1));
      wait_async<CHUNK>();   // chunk c landed; chunk c+1 still in flight
    } else {
      wait_async<0>();
    }
    const float2* lb = &buf[c & 1][tid];
    const int t0 = c * CHUNK;
    const int n  = min(CHUNK, T_TOTAL - t0);

#pragma unroll 4
    for (int k = 0; k < n; ++k) {
      float2 pe2 = lb[k * BLOCK];          // ds_load_b64, wave-private slot
      float p = fmaxf(pe2.x, 0.0f);
      float e = fmaxf(pe2.y, 0.0f);

      // --- runoff generation (water balance) ---
      float ep = kc * e;
      float et = fminf(ep, p + xp + xf);
      float pe = fmaxf(p - et, 0.0f);
      float x  = xf;
      xf = fmaxf(x - fmaxf(ep - p, 0.0f), 0.0f);
      xp = fmaxf(xp - fmaxf(ep - p - x, 0.0f), 0.0f);
      float t1 = fmaxf(k1 * fminf(x2, w1 - xp), 0.0f);
      xp += t1;
      x2 = fmaxf(x2 - t1, 0.0f);
      float t2 = fmaxf(kw * (xs * w1 - xp * w2), 0.0f);
      xp += t2;
      xs = fmaxf(xs - t2, 0.0f);
      xf += fmaxf(xp + pe - w1, 0.0f);
      xp = fminf(w1, xp + pe);
      float f1 = a0 * xf;
      float rs = (xf > h2) ? fmaf(xf - h2, a2, (xf - h1) * a1)
                           : ((xf > h1) ? (xf - h1) * a1 : 0.0f);
      x2 += f1;
      xf = fmaxf(xf - (rs + f1), 0.0f);
      float f2 = b0 * x2;
      float ri = (x2 > h3) ? (x2 - h3) * b1 : 0.0f;
      x2 = fmaxf(x2 - (f2 + ri), 0.0f);
      x3 += f2;
      float f3 = c0 * x3;
      float rgs = (x3 > h4) ? (x3 - h4) * c1 : 0.0f;
      x3 = fmaxf(x3 - (f3 + rgs), 0.0f);
      x4 += f3;
      float rgd = d1 * x4;
      x4 = fmaxf(x4 - rgd, 0.0f);

      // --- routing (division-free: precomputed reciprocals + selects) ---
      bool  hi0 = (x5 >= h);
      float kk  = hi0 ? kb : ka;           // kk0 - dt
      float cc0 = hi0 ? ch : 0.0f;
      float ii  = (rs + ri) + (rgs + rgd);
      float q1  = fmaxf(fmaf(kk, qs, ii + cc0) * da, 0.0f);
      x5 = inv_e1 * q1;
      bool  hi1  = (x5 > h);
      float rc1  = hi1 ? ch : 0.0f;
      float dsel = hi1 ? db : da;
      q1 = fmaxf(fmaf(kk, qs, (ii - rc1) + cc0) * dsel, 0.0f);
      x5 = fmaf(hi1 ? inv_e12 : inv_e1, q1, rc1);
      qs = q1;

      int t = t0 + k;                      // wave-uniform -> scalar branch
      if (t >= TWARM) {
        size_t idx = (size_t)(t - TWARM) * B + b;
        __builtin_nontemporal_store(q1, qout + idx);   // write-once: bypass cache
        __builtin_nontemporal_store(et, eout + idx);
      }
    }
  }
}

extern "C" void kernel_launch(void* const* d_in, const int* in_sizes, int n_in,
                              void* d_out, int out_size, void* d_ws, size_t ws_size,
                              hipStream_t stream) {
  const float* p_and_e = (const float*)d_in[0];   // [1096, B, 2] f32
  const float* params  = (const float*)d_in[1];   // [B, 20] f32
  float* out = (float*)d_out;                     // q[731*B] ++ e[731*B]
  const int B = in_sizes[1] / 20;                 // 16384
  const int grid = (B + BLOCK - 1) / BLOCK;       // 256 blocks x 64 threads
  hipLaunchKernelGGL(tank_scan_kernel, dim3(grid), dim3(BLOCK), 0, stream,
                     p_and_e, params, out, B);
}